// Attention_47493748359701
// MI455X (gfx1250) — compile-verified
//
#include <hip/hip_runtime.h>
#include <hip/hip_bf16.h>
#include <math.h>

#define BB 32
#define TT 4096
#define EE 300
#define BT (BB * TT)          // 131072 rows
#define KSTEPS (EE / 4)       // 75 exact f32-WMMA K-steps

typedef __attribute__((ext_vector_type(2))) float v2f;
typedef __attribute__((ext_vector_type(4))) float v4f;
typedef __attribute__((ext_vector_type(8))) float v8f;

__device__ __forceinline__ float fast_tanh(float x) {
#if __has_builtin(__builtin_amdgcn_tanhf)
    return __builtin_amdgcn_tanhf(x);   // v_tanh_f32 (gfx1250 tanh-insts)
#else
    return tanhf(x);
#endif
}
__device__ __forceinline__ float fast_sigmoid(float x) {
    // sigmoid(x) = 0.5*tanh(0.5x) + 0.5 : single v_tanh_f32 on critical path
    return fmaf(0.5f, fast_tanh(0.5f * x), 0.5f);
}

// ---------------------------------------------------------------------------
// Kernel 1: xg[row][n] = x[row,:] . w_pad[n,:] + bias[n]   (n = 0..7 real)
// f32 WMMA 16x16x4: M-tile = 16 rows of flattened (b*T+t), N = 16 (8 used),
// K = 300 = 75 * 4. 128 threads = 4 waves, 1 M-tile per wave.
// x is read with default (RT) temporal hint so it stays L2-resident for the
// final scale kernel (x = 157 MB < 192 MB global L2).
// ---------------------------------------------------------------------------
__global__ void gates_wmma_kernel(const float* __restrict__ x,
                                  const float* __restrict__ w_ih_f,
                                  const float* __restrict__ w_ih_b,
                                  const float* __restrict__ b_ih_f,
                                  const float* __restrict__ b_hh_f,
                                  const float* __restrict__ b_ih_b,
                                  const float* __restrict__ b_hh_b,
                                  float* __restrict__ xg) {
    __shared__ float wlds[16][304];  // padded 16x300 weight matrix (B operand)

    for (int i = threadIdx.x; i < 16 * 304; i += blockDim.x) {
        int n = i / 304, k = i - n * 304;
        float v = 0.0f;
        if (k < EE) {
            if (n < 4)      v = w_ih_f[n * EE + k];
            else if (n < 8) v = w_ih_b[(n - 4) * EE + k];
        }
        wlds[n][k] = v;
    }
    __syncthreads();

    const int wave  = threadIdx.x >> 5;   // wave32
    const int lane  = threadIdx.x & 31;
    const int tile  = blockIdx.x * 4 + wave;
    const int row0  = tile * 16;
    const int m     = lane & 15;          // A: row within tile / B: column
    const int khalf = lane >> 4;          // 0 or 1 -> K sub-offset 0 or 2

    const float* arow = x + (size_t)(row0 + m) * EE + 2 * khalf;
    const float* wrow = &wlds[m][2 * khalf];

    v8f acc = {};
#pragma unroll 5
    for (int kk = 0; kk < EE; kk += 4) {
        v2f a = *(const v2f*)(arow + kk);   // global_load_b64
        v2f b = *(const v2f*)(wrow + kk);   // ds_load_b64
        acc = __builtin_amdgcn_wmma_f32_16x16x4_f32(
            /*neg_a=*/false, a, /*neg_b=*/false, b,
            /*c_mod=*/(short)0, acc, /*reuse_a=*/false, /*reuse_b=*/false);
    }

    const int n = lane & 15;
    if (n < 8) {
        const int dir = n >> 2, g = n & 3;
        const float bias = (dir == 0) ? (b_ih_f[g] + b_hh_f[g])
                                      : (b_ih_b[g] + b_hh_b[g]);
        const int rbase = row0 + 8 * khalf;   // C/D: VGPR v -> row v + 8*khalf
#pragma unroll
        for (int v = 0; v < 8; ++v) {
            xg[(size_t)(rbase + v) * 8 + n] = acc[v] + bias;
        }
    }
}

// ---------------------------------------------------------------------------
// Kernel 2: 64 independent LSTM chains (32 batches x 2 directions), one lane
// each. Latency-bound: the next step's gates are preloaded BEFORE the current
// step's serial tanh chain so the global-load latency overlaps compute, and
// prefetch (global_prefetch_b8) runs ahead one cacheline every 4 steps.
// ---------------------------------------------------------------------------
__global__ void lstm_scan_kernel(const float* __restrict__ xg,
                                 const float* __restrict__ w_hh_f,
                                 const float* __restrict__ w_hh_b,
                                 float* __restrict__ hf,
                                 float* __restrict__ hb) {
    const int l   = threadIdx.x;      // 0..63
    const int dir = l >> 5;
    const int b   = l & 31;
    const float* wh = dir ? w_hh_b : w_hh_f;   // [4H,1] -> 4 scalars
    const float w0 = wh[0], w1 = wh[1], w2 = wh[2], w3 = wh[3];
    float* __restrict__ hout = dir ? hb : hf;

    const float* base = xg + (size_t)b * TT * 8 + 4 * dir;
    float h = 0.0f, c = 0.0f;

    const int t0 = dir ? (TT - 1) : 0;
    v4f g4 = *(const v4f*)(base + (size_t)t0 * 8);

    for (int s = 0; s < TT; ++s) {
        const int t  = dir ? (TT - 1 - s) : s;
        const int sn = (s + 1 < TT) ? (s + 1) : s;
        const int tn = dir ? (TT - 1 - sn) : sn;

        // issue next step's gate load before this step's dependent chain
        const v4f gnext = *(const v4f*)(base + (size_t)tn * 8);

        if ((s & 3) == 0) {
            const int tp = dir ? (t - 32) : (t + 32);
            if (tp >= 0 && tp < TT)
                __builtin_prefetch(base + (size_t)tp * 8, 0, 0);
        }

        const float gi = fmaf(h, w0, g4.x);
        const float gf = fmaf(h, w1, g4.y);
        const float gc = fmaf(h, w2, g4.z);
        const float go = fmaf(h, w3, g4.w);
        const float si = fast_sigmoid(gi);
        const float sf = fast_sigmoid(gf);
        const float so = fast_sigmoid(go);
        const float tg = fast_tanh(gc);
        c = fmaf(sf, c, si * tg);
        h = so * fast_tanh(c);
        hout[(size_t)b * TT + t] = h;

        g4 = gnext;
    }
}

// ---------------------------------------------------------------------------
// Kernel 3: att[b,:] = softmax_t( 0.5*(hf[b,t]+hb[b,t]) ). One block per b.
// ---------------------------------------------------------------------------
__global__ void attn_softmax_kernel(const float* __restrict__ hf,
                                    const float* __restrict__ hb,
                                    float* __restrict__ att) {
    const int b = blockIdx.x;
    __shared__ float red[256];
    const float* f  = hf + (size_t)b * TT;
    const float* bk = hb + (size_t)b * TT;

    float mx = -INFINITY;
    for (int t = threadIdx.x; t < TT; t += 256)
        mx = fmaxf(mx, 0.5f * (f[t] + bk[t]));
    red[threadIdx.x] = mx;
    __syncthreads();
    for (int s = 128; s > 0; s >>= 1) {
        if (threadIdx.x < s)
            red[threadIdx.x] = fmaxf(red[threadIdx.x], red[threadIdx.x + s]);
        __syncthreads();
    }
    mx = red[0];
    __syncthreads();

    float sum = 0.0f;
    for (int t = threadIdx.x; t < TT; t += 256)
        sum += __expf(0.5f * (f[t] + bk[t]) - mx);
    red[threadIdx.x] = sum;
    __syncthreads();
    for (int s = 128; s > 0; s >>= 1) {
        if (threadIdx.x < s)
            red[threadIdx.x] += red[threadIdx.x + s];
        __syncthreads();
    }
    const float inv = 1.0f / red[0];

    for (int t = threadIdx.x; t < TT; t += 256)
        att[(size_t)b * TT + t] = __expf(0.5f * (f[t] + bk[t]) - mx) * inv;
}

// ---------------------------------------------------------------------------
// Kernel 4: out[b,t,e] = att[b,t] * x[b,t,e].  v4f streaming; E%4==0 so each
// 4-vector lies in one row; row = i4 / 75.  x read normally (should hit L2
// from kernel 1); out written NON-TEMPORAL (TH=NT) so the 157 MB of write
// data does not evict x from the 192 MB L2.
// ---------------------------------------------------------------------------
__global__ void scale_kernel(const float* __restrict__ x,
                             const float* __restrict__ att,
                             float* __restrict__ out) {
    const unsigned i4 = blockIdx.x * 256u + threadIdx.x;
    const unsigned n4 = (unsigned)BT * (EE / 4);   // 9,830,400
    if (i4 >= n4) return;
    const unsigned row = i4 / (EE / 4);
    const float a = att[row];
    v4f v = ((const v4f*)x)[i4];
    v *= a;
    __builtin_nontemporal_store(v, (v4f*)out + i4);   // TH=NT store
}

// ---------------------------------------------------------------------------
extern "C" void kernel_launch(void* const* d_in, const int* in_sizes, int n_in,
                              void* d_out, int out_size, void* d_ws, size_t ws_size,
                              hipStream_t stream) {
    const float* x      = (const float*)d_in[0];
    const float* w_ih_f = (const float*)d_in[1];
    const float* w_hh_f = (const float*)d_in[2];
    const float* b_ih_f = (const float*)d_in[3];
    const float* b_hh_f = (const float*)d_in[4];
    const float* w_ih_b = (const float*)d_in[5];
    const float* w_hh_b = (const float*)d_in[6];
    const float* b_ih_b = (const float*)d_in[7];
    const float* b_hh_b = (const float*)d_in[8];
    float* out = (float*)d_out;

    float* xg  = (float*)d_ws;              // [BT][8]  = 4 MB
    float* hf  = xg + (size_t)BT * 8;       // [B][T]
    float* hb  = hf + (size_t)BT;           // [B][T]
    float* att = hb + (size_t)BT;           // [B][T]

    // 1) Fused 2-direction input GEMM via f32 WMMA: 8192 M-tiles, 4 per block
    gates_wmma_kernel<<<dim3(BT / 16 / 4), dim3(128), 0, stream>>>(
        x, w_ih_f, w_ih_b, b_ih_f, b_hh_f, b_ih_b, b_hh_b, xg);

    // 2) 64 sequential LSTM chains
    lstm_scan_kernel<<<dim3(1), dim3(64), 0, stream>>>(xg, w_hh_f, w_hh_b, hf, hb);

    // 3) Softmax over T per batch
    attn_softmax_kernel<<<dim3(BB), dim3(256), 0, stream>>>(hf, hb, att);

    // 4) Broadcast-scale x
    const unsigned n4 = (unsigned)BT * (EE / 4);
    scale_kernel<<<dim3((n4 + 255) / 256), dim3(256), 0, stream>>>(x, att, out);
}